// VolumeRenderer_9105330667607
// MI455X (gfx1250) — compile-verified
//
#include <hip/hip_runtime.h>
#include <stdint.h>

// ---------------------------------------------------------------------------
// NeRF volume rendering (memory-bound scan, HBM floor ~61us @ 23.3 TB/s).
// MI455X strategy:
//   - TDM (tensor_load_to_lds) stages each wave's 32-ray x 16-sample radiance
//     tile into wave-private, double-buffered LDS (TENSORcnt pipelined, no
//     workgroup barriers needed).
//   - LDS rows padded to 272B so ds_load_b128 is bank-conflict free.
//   - 69.6KB LDS/WG -> 4 workgroups per 320KB WGP (4 waves/SIMD).
//   - Depth read as aligned float4 with one-chunk lookahead; chunk-final dist
//     coincides with the 1e10 sentinel at s=191.
//   - weights/mask are write-once streams -> nontemporal b128 stores.
// ---------------------------------------------------------------------------

#define NS         192
#define BLOCK      128            // 4 waves of 32 (wave32)
#define RPW        32             // rays per wave
#define S_TILE     16             // samples per tile
#define NTILES     (NS / S_TILE)  // 12
#define ROW_BYTES  272            // 16 samples * 16B + 16B TDM pad
#define WBUF_BYTES (RPW * ROW_BYTES)   // 8704
#define WAVE_BYTES (2 * WBUF_BYTES)    // 17408 (double buffer)
#define LDS_TOTAL  (4 * WAVE_BYTES)    // 69632  -> 4 WGs / 320KB WGP

#define ONE_E_10 1.0e10f
#define ATT_TH   0.001f
#define EPSF     1.0e-10f

typedef __attribute__((ext_vector_type(4))) float        f32x4;
typedef __attribute__((ext_vector_type(4))) unsigned int tdm_v4u;
typedef __attribute__((ext_vector_type(8))) int          tdm_v8i;
typedef __attribute__((ext_vector_type(4))) int          tdm_v4i;

// 2D TDM tile load: 32 x 8B units per row (16 float4 samples), 32 rows (rays),
// global row-to-row stride 384 x 8B units (= 192 samples * 16B).
// LDS padding: +16B after every 256B  => 272B LDS row stride.
__device__ __forceinline__ void tdm_load_tile(unsigned lds_byte_addr,
                                              unsigned long long gaddr)
{
    tdm_v4u g0;
    g0.x = 1u;                                        // count=1, user descriptor
    g0.y = lds_byte_addr;                             // LDS dest (bytes)
    g0.z = (unsigned)(gaddr & 0xFFFFFFFFull);         // global addr [31:0]
    g0.w = (unsigned)((gaddr >> 32) & 0x1FFFFFFull)   // global addr [56:32]
         | (2u << 30);                                // type = 2 ("image")

    const unsigned td0 = (1u << 30);  // tensor_dim0: large -> no OOB clipping
    const unsigned td1 = (1u << 30);  // tensor_dim1: large -> no OOB clipping
    tdm_v8i g1;
    g1[0] = (int)((3u << 16)      // data_size = 8B units
                | (1u << 20)      // pad_enable
                | (5u << 22)      // pad_interval: 2^5 x 8B = 256B
                | (3u << 25));    // pad_amount: 4 DWORDs = 16B
    g1[1] = (int)((td0 & 0xFFFFu) << 16);                       // tensor_dim0 lo16
    g1[2] = (int)((td0 >> 16) | ((td1 & 0xFFFFu) << 16));       // dim0 hi / dim1 lo
    g1[3] = (int)(((td1 >> 16) & 0xFFFFu) | (32u << 16));       // dim1 hi / tile_dim0=32
    g1[4] = (int)32u;                                           // tile_dim1=32, tile_dim2=0
    g1[5] = (int)384;                                           // tensor_dim0_stride lo (8B units)
    g1[6] = 0;                                                  // stride hi / dim1_stride lo
    g1[7] = 0;

    tdm_v4i gz = {0, 0, 0, 0};   // groups 2/3 unused (2D tile)
#if defined(__clang_major__) && (__clang_major__ >= 23)
    tdm_v8i gz8 = {0, 0, 0, 0, 0, 0, 0, 0};
    __builtin_amdgcn_tensor_load_to_lds(g0, g1, gz, gz, gz8, 0);
#else
    __builtin_amdgcn_tensor_load_to_lds(g0, g1, gz, gz, 0);
#endif
}

__global__ void __launch_bounds__(BLOCK)
nerf_volume_render(const float* __restrict__ rf,    // (n,192,4)
                   const float* __restrict__ depth, // (n,192)
                   const float* __restrict__ rdir,  // (n,3)
                   float* __restrict__ out,
                   int n_rays)
{
    __shared__ __align__(1024) unsigned char smem[LDS_TOTAL];

    const int tid  = threadIdx.x;
    const int wave = tid >> 5;
    const int lane = tid & 31;
    const int ray  = blockIdx.x * BLOCK + tid;
    const int wray0 = blockIdx.x * BLOCK + wave * RPW;  // wave's first ray

    // Low 32 bits of a generic LDS pointer == LDS byte offset on gfx1250.
    const unsigned lds_wave = (unsigned)(uintptr_t)(&smem[0])
                            + (unsigned)(wave * WAVE_BYTES);

    const unsigned long long rf_wave =
        (unsigned long long)(uintptr_t)rf + (unsigned long long)wray0 * (NS * 16ull);

    // Kick off tile 0 for this wave before any other work.
    tdm_load_tile(lds_wave, rf_wave);

    // Per-ray setup (overlaps with the DMA).
    const float dx = rdir[(size_t)ray * 3 + 0];
    const float dy = rdir[(size_t)ray * 3 + 1];
    const float dz = rdir[(size_t)ray * 3 + 2];
    const float nrm = sqrtf(dx * dx + dy * dy + dz * dz);

    const float* drow = depth + (size_t)ray * NS;
    __builtin_prefetch(drow, 0, 1);   // global_prefetch_b8 for the depth row

    float T = 1.0f, accw = 0.0f, dacc = 0.0f;
    float cr = 0.0f, cg = 0.0f, cb = 0.0f;

    float* out_rgb  = out;
    float* out_dep  = out + (size_t)3   * n_rays;
    float* out_w    = out + (size_t)4   * n_rays;
    float* out_m    = out + (size_t)196 * n_rays;
    float* out_acc  = out + (size_t)388 * n_rays;
    float* out_disp = out + (size_t)389 * n_rays;

    f32x4 D = *(const f32x4*)(drow);   // depth[0..3]

    for (int t = 0; t < NTILES; ++t) {
        if (t + 1 < NTILES) {
            // Prefetch next tile into the other half of the wave's buffer,
            // then wait for the *older* outstanding TDM op (tile t).
            tdm_load_tile(lds_wave + (unsigned)(((t + 1) & 1) * WBUF_BYTES),
                          rf_wave + (unsigned long long)(t + 1) * (S_TILE * 16ull));
            __builtin_amdgcn_s_wait_tensorcnt((short)1);
        } else {
            __builtin_amdgcn_s_wait_tensorcnt((short)0);
        }

        // Wave-private data: no workgroup barrier needed.
        const unsigned char* rowp = smem + (size_t)wave * WAVE_BYTES
                                  + (size_t)((t & 1) * WBUF_BYTES)
                                  + (size_t)lane * ROW_BYTES;

        for (int cl = 0; cl < S_TILE / 4; ++cl) {
            const int c = t * (S_TILE / 4) + cl;      // global 4-sample chunk
            f32x4 Dn = D;
            const bool has_next = (c < NS / 4 - 1);
            if (has_next) Dn = *(const f32x4*)(drow + 4 * (c + 1));

            float dsmp[4] = { D.x, D.y, D.z, D.w };
            float dist[4];
            dist[0] = (D.y - D.x) * nrm;
            dist[1] = (D.z - D.y) * nrm;
            dist[2] = (D.w - D.z) * nrm;
            dist[3] = has_next ? (Dn.x - D.w) * nrm   // only missing at s=191,
                               : ONE_E_10 * nrm;      // which is the sentinel

            float wk[4], mk[4];
#pragma unroll
            for (int k = 0; k < 4; ++k) {
                const f32x4 v = *(const f32x4*)(rowp + (size_t)(cl * 4 + k) * 16);

                const float sigma = fmaxf(v.w, 0.0f);
                const float alpha = 1.0f - __expf(-sigma * dist[k]);
                const float w     = alpha * T;

                mk[k] = (T > ATT_TH) ? 1.0f : 0.0f;
                wk[k] = w;

                cr   += w * v.x;
                cg   += w * v.y;
                cb   += w * v.z;
                accw += w;
                dacc += w * dsmp[k];

                T = T * (1.0f - alpha + EPSF);
            }

            const size_t o = (size_t)ray * NS + (size_t)c * 4;
            f32x4 wv = { wk[0], wk[1], wk[2], wk[3] };
            f32x4 mv = { mk[0], mk[1], mk[2], mk[3] };
            __builtin_nontemporal_store(wv, (f32x4*)(out_w + o));
            __builtin_nontemporal_store(mv, (f32x4*)(out_m + o));

            D = Dn;
        }
    }

    out_rgb[(size_t)ray * 3 + 0] = cr;
    out_rgb[(size_t)ray * 3 + 1] = cg;
    out_rgb[(size_t)ray * 3 + 2] = cb;
    out_acc[ray] = accw;

    const float q = dacc / accw;                 // may be NaN (0/0)
    float disp = isnan(q) ? 0.0f : (1.0f / fmaxf(EPSF, q));
    if (isnan(disp)) disp = 0.0f;
    out_disp[ray] = disp;
    out_dep[ray]  = (accw < 1.0f) ? 0.0f : dacc;
}

extern "C" void kernel_launch(void* const* d_in, const int* in_sizes, int n_in,
                              void* d_out, int out_size, void* d_ws, size_t ws_size,
                              hipStream_t stream) {
    const float* rf    = (const float*)d_in[0];  // radiance_field (n,192,4)
    const float* depth = (const float*)d_in[1];  // depth_values   (n,192)
    const float* rdir  = (const float*)d_in[2];  // ray_directions (n,3)
    float* out = (float*)d_out;

    const int n_rays = in_sizes[2] / 3;          // 262144
    dim3 grid((unsigned)(n_rays / BLOCK)), block(BLOCK);
    hipLaunchKernelGGL(nerf_volume_render, grid, block, 0, stream,
                       rf, depth, rdir, out, n_rays);
}